// IouSimilarity_61546881352239
// MI455X (gfx1250) — compile-verified
//
#include <hip/hip_runtime.h>
#include <cstdint>
#include <cstddef>

#ifndef __has_builtin
#define __has_builtin(x) 0
#endif

#define MASK_VAL (-1.0f)

// Native clang vectors (HIP float4 is a struct and won't feed the builtins)
typedef float v4f __attribute__((ext_vector_type(4)));
typedef int   v4i __attribute__((vector_size(16)));   // matches builtin param pointee

// ---- CDNA5 async global->LDS path (guarded, falls back to plain staging) ----
#if defined(__gfx1250__) && __has_builtin(__builtin_amdgcn_global_load_async_to_lds_b128)
#define USE_ASYNC_LDS 1
typedef __attribute__((address_space(1))) v4i* as1_v4i;   // global src
typedef __attribute__((address_space(3))) v4i* as3_v4i;   // LDS dst
#else
#define USE_ASYNC_LDS 0
#endif

__device__ __forceinline__ void wait_async_zero() {
#if defined(__gfx1250__)
#if __has_builtin(__builtin_amdgcn_s_wait_asynccnt)
    __builtin_amdgcn_s_wait_asynccnt(0);
#else
    asm volatile("s_wait_asynccnt 0" ::: "memory");
#endif
#endif
}

// Tile geometry: block covers BI rows x BJ cols of the [N x M] IoU matrix.
// 256 threads (8 wave32), each computing a 4x8 micro-tile -> 32 x 256 block tile.
#define BI 32
#define BJ 256

__global__ __launch_bounds__(256) void iou_similarity_kernel(
    const float* __restrict__ boxes1,  // [B, N, 4]
    const float* __restrict__ boxes2,  // [B, M, 4]
    float* __restrict__ out,           // [B, N, M]
    int N, int M)
{
    __shared__ v4f sA[BI];   // boxes1 tile (32 boxes, 512 B)
    __shared__ v4f sB[BJ];   // boxes2 tile (256 boxes, 4 KB)

    const int b   = blockIdx.z;
    const int i0  = blockIdx.y * BI;
    const int j0  = blockIdx.x * BJ;
    const int tid = threadIdx.x;

    const v4f* gA = (const v4f*)(boxes1 + (size_t)b * N * 4) + i0;
    const v4f* gB = (const v4f*)(boxes2 + (size_t)b * M * 4) + j0;

    // ---- Stage box tiles into LDS (async DMA on CDNA5) ----
#if USE_ASYNC_LDS
    __builtin_amdgcn_global_load_async_to_lds_b128(
        (as1_v4i)(gB + tid), (as3_v4i)(sB + tid), 0, 0);
    if (tid < BI) {
        __builtin_amdgcn_global_load_async_to_lds_b128(
            (as1_v4i)(gA + tid), (as3_v4i)(sA + tid), 0, 0);
    }
    wait_async_zero();
#else
    sB[tid] = gB[tid];
    if (tid < BI) sA[tid] = gA[tid];
#endif
    __syncthreads();

    // ---- 4x8 register micro-tile per thread ----
    const int tj = (tid & 31) << 3;   // 0..248, column offset within tile
    const int ti = (tid >> 5) << 2;   // 0..28,  row offset within tile

    v4f A[4];  float area1[4];  bool m1[4];
#pragma unroll
    for (int r = 0; r < 4; ++r) {
        A[r] = sA[ti + r];
        area1[r] = (A[r].z - A[r].x) * (A[r].w - A[r].y);
        m1[r] = fmaxf(fmaxf(A[r].x, A[r].y), fmaxf(A[r].z, A[r].w)) < 0.0f;
    }

    v4f Bb[8]; float area2[8];  bool m2[8];
#pragma unroll
    for (int c = 0; c < 8; ++c) {
        Bb[c] = sB[tj + c];
        area2[c] = (Bb[c].z - Bb[c].x) * (Bb[c].w - Bb[c].y);
        m2[c] = fmaxf(fmaxf(Bb[c].x, Bb[c].y), fmaxf(Bb[c].z, Bb[c].w)) < 0.0f;
    }

    // Single 64-bit base; the 8 stores are base + r * rowStride (runtime M)
    // + half; only the 16 B half step is a compile-time constant.
    v4f* const dst0 = (v4f*)(out + ((size_t)(b * N + i0 + ti)) * M + (j0 + tj));
    const int rowStride = M >> 2;     // v4f elements per output row (1024)

#pragma unroll
    for (int r = 0; r < 4; ++r) {
#pragma unroll
        for (int half = 0; half < 2; ++half) {
            float res[4];
#pragma unroll
            for (int q = 0; q < 4; ++q) {
                const int c = half * 4 + q;
                const float ltx = fmaxf(A[r].x, Bb[c].x);
                const float lty = fmaxf(A[r].y, Bb[c].y);
                const float rbx = fminf(A[r].z, Bb[c].z);
                const float rby = fminf(A[r].w, Bb[c].w);
                const float w   = fmaxf(rbx - ltx, 0.0f);
                const float h   = fmaxf(rby - lty, 0.0f);
                const float inter = w * h;
                const float uni   = area1[r] + area2[c] - inter;
                // divide_no_nan semantics: 0 where union <= 0
                float iou = (uni > 0.0f) ? inter * __builtin_amdgcn_rcpf(uni) : 0.0f;
                if (m1[r] || m2[c]) iou = MASK_VAL;
                res[q] = iou;
            }
            v4f v = {res[0], res[1], res[2], res[3]};
            __builtin_nontemporal_store(v, dst0 + r * rowStride + half);
        }
    }
}

extern "C" void kernel_launch(void* const* d_in, const int* in_sizes, int n_in,
                              void* d_out, int out_size, void* d_ws, size_t ws_size,
                              hipStream_t stream) {
    (void)n_in; (void)d_ws; (void)ws_size;
    const float* boxes1 = (const float*)d_in[0];   // [4, 4096, 4] f32
    const float* boxes2 = (const float*)d_in[1];   // [4, 4096, 4] f32
    float* out = (float*)d_out;                    // [4, 4096, 4096] f32

    constexpr int B = 4;
    constexpr int N = 4096;
    constexpr int M = 4096;
    (void)in_sizes; (void)out_size;

    dim3 grid(M / BJ, N / BI, B);   // (16, 128, 4) = 8192 blocks
    iou_similarity_kernel<<<grid, 256, 0, stream>>>(boxes1, boxes2, out, N, M);
}